// BlockMLP_60318520705580
// MI455X (gfx1250) — compile-verified
//
#include <hip/hip_runtime.h>

#define V_      8
#define N_      128
#define TOT_    20480
#define Z_      2
#define NCHR_   10
#define BPC_    8
#define BLK_    256
#define NB_     80
#define SLICES_ 5
#define BPS_    16     // blocks per slice (2 chromosomes)
#define STRX_   516    // padded LDS x-row stride in floats (512 data + 4 pad)
#define WSTR_   516    // stride between the two pre-masked W1 halves

typedef float v2f __attribute__((ext_vector_type(2)));
typedef float v8f __attribute__((ext_vector_type(8)));

// Async global->LDS copy (CDNA5 async path, ASYNCcnt-tracked).
// Per CDNA5 ISA: copies global memory into LDS (GV mode),
// the 24-bit immediate offset applies to BOTH addresses.
#define ASYNC_CP_B128(LA, GA, IMM)                                          \
    asm volatile("global_load_async_to_lds_b128 %0, %1, off offset:" #IMM  \
                 :: "v"(LA), "v"(GA) : "memory")

// ---------------------------------------------------------------------------
// Kernel 1: layer-1 dots via V_WMMA_F32_16X16X4_F32 over the raw interleaved
// (t,z) axis. z-separation lives in the B columns: even columns see the
// pre-masked (W1[t],0) interleave, odd columns see (0,W1[t]), so ONE wmma
// chain yields both ploidy dots with zero VALU in the loop. Softmax-pool
// over z, per-chromosome layer-2, slice partials -> workspace.
// Workgroup = (v, 16-n tile, chromosome-pair slice); 256 threads = 8 waves.
// ---------------------------------------------------------------------------
__global__ __launch_bounds__(256) void blockmlp_l1(
    const float* __restrict__ x,  const float* __restrict__ W1,
    const float* __restrict__ b1, const float* __restrict__ W2,
    const float* __restrict__ b2, const float* __restrict__ Wa,
    const float* __restrict__ Wc,
    float* __restrict__ wsA, float* __restrict__ wsC)
{
    __shared__ float lds_x[16 * STRX_];    // raw tile: [m][k' = t*2+z], padded
    __shared__ float lds_wint[2 * WSTR_];  // pre-masked interleaved W1 halves
    __shared__ float y_part[8 * 2 * 16];   // [wave][z][m] partial dots

    const int tid  = threadIdx.x;
    const int nt   = blockIdx.x;    // n-tile (16 n's)
    const int v    = blockIdx.y;
    const int s    = blockIdx.z;    // slice: chromosomes {2s, 2s+1}
    const int n0   = nt * 16;

    const int lane = tid & 31;
    const int wv   = tid >> 5;          // wave id 0..7 -> k'-eighth
    const int mloc = lane & 15;         // WMMA M row
    const int hi   = lane >> 4;         // 0: K+0..1, 1: K+2..3
    const int koff = hi << 1;
    const int jodd = lane & 1;          // B/D column parity == z

    const int mrow = tid >> 4;          // staging: n-row 0..15
    const int sub  = tid & 15;          // staging: 16 threads per row

    float acc0 = 0.f, acc1 = 0.f;       // layer-2 accumulators (threads 0..15)

    const float* rowp =
        x + ((long)(v * N_ + n0 + mrow) * TOT_ + (long)s * BPS_ * BLK_) * Z_;

    // invariant per-lane LDS read bases for the WMMA loop
    const float* arow  = &lds_x[mloc * STRX_];
    const float* wbase = &lds_wint[jodd * WSTR_ + koff];

    for (int bl = 0; bl < BPS_; ++bl) {
        const int b = s * BPS_ + bl;

        // ---- stage pre-masked interleaved W1 row (2 x 512 floats) ----
        {
            float w = W1[b * BLK_ + tid];
            *(float2*)(&lds_wint[2 * tid])         = make_float2(w, 0.f);
            *(float2*)(&lds_wint[WSTR_ + 2 * tid]) = make_float2(0.f, w);
        }

        // ---- async-stage raw x tile: 16 n x 512 floats = 32 KB ----
        const float* rp = rowp + (long)bl * BLK_ * Z_;
        {
            unsigned long long ga =
                (unsigned long long)(const void*)(rp + sub * 4);       // +16B*sub
            unsigned int la = (unsigned int)(unsigned long long)
                (const void*)&lds_x[mrow * STRX_ + sub * 4];
            ASYNC_CP_B128(la, ga, 0);
            ASYNC_CP_B128(la, ga, 256);
            ASYNC_CP_B128(la, ga, 512);
            ASYNC_CP_B128(la, ga, 768);
            ASYNC_CP_B128(la, ga, 1024);
            ASYNC_CP_B128(la, ga, 1280);
            ASYNC_CP_B128(la, ga, 1536);
            ASYNC_CP_B128(la, ga, 1792);
        }
        // prefetch next block's tile (global_prefetch_b8)
        if (bl + 1 < BPS_ && (sub & 3) == 0) {
            #pragma unroll
            for (int i = 0; i < 8; ++i)
                __builtin_prefetch(rp + BLK_ * Z_ + sub * 4 + i * 64, 0, 1);
        }
        asm volatile("s_wait_asynccnt 0x0" ::: "memory");
        __syncthreads();

        // ---- WMMA: this wave covers k' in [wv*64, wv*64+64) ----
        v8f c = {0.f, 0.f, 0.f, 0.f, 0.f, 0.f, 0.f, 0.f};
        #pragma unroll
        for (int i = 0; i < 16; ++i) {
            const int kk = wv * 64 + i * 4;                 // chunk base (mult of 4)
            v2f a  = *(const v2f*)(&arow[kk + koff]);       // A: x[m, k'..k'+1]
            v2f bf = *(const v2f*)(&wbase[kk]);             // pre-masked W1 pair
            c = __builtin_amdgcn_wmma_f32_16x16x4_f32(
                    false, a, false, bf, (short)0, c, false, false);
        }
        // D: even columns = z0 dot, odd columns = z1 dot.
        // lanes 0/1 hold rows 0-7 (N=0/1), lanes 16/17 hold rows 8-15.
        if (mloc < 2) {
            const int zsel = mloc;
            const int mb = hi * 8;
            #pragma unroll
            for (int r = 0; r < 8; ++r)
                y_part[(wv * 2 + zsel) * 16 + mb + r] = c[r];
        }
        __syncthreads();

        // ---- fixed-order wave-sum, bias+relu, softmax-pool, layer-2 ----
        if (tid < 16) {
            const int m = tid;
            float y0 = 0.f, y1 = 0.f;
            #pragma unroll
            for (int w = 0; w < 8; ++w) {
                y0 += y_part[(w * 2 + 0) * 16 + m];
                y1 += y_part[(w * 2 + 1) * 16 + m];
            }
            float bb = b1[b];
            y0 = fmaxf(y0 + bb, 0.f);
            y1 = fmaxf(y1 + bb, 0.f);
            float mx = fmaxf(y0, y1);
            float e0 = __expf(y0 - mx), e1 = __expf(y1 - mx);
            float pooled = (e0 * y0 + e1 * y1) / (e0 + e1);
            float w2v = W2[(b >> 3) * BPC_ + (b & 7)];
            if (bl < 8) acc0 += pooled * w2v; else acc1 += pooled * w2v;
        }
        __syncthreads();   // protect lds_x / lds_wint / y_part before next iter
    }

    if (tid < 16) {
        const int c0 = 2 * s, c1 = 2 * s + 1;
        float f0 = fmaxf(acc0 + b2[c0], 0.f);
        float f1 = fmaxf(acc1 + b2[c1], 0.f);
        const int idx = (s * V_ + v) * N_ + n0 + tid;
        wsA[idx] = f0 * Wa[c0] + f1 * Wa[c1];
        wsC[idx] = f0 * Wc[c0] + f1 * Wc[c1];
    }
}

// ---------------------------------------------------------------------------
// Kernel 2: deterministic slice-sum + head biases + critic mean over N.
// ---------------------------------------------------------------------------
__global__ __launch_bounds__(128) void blockmlp_head(
    const float* __restrict__ wsA, const float* __restrict__ wsC,
    const float* __restrict__ ba,  const float* __restrict__ bc,
    float* __restrict__ out)
{
    __shared__ float red[128];
    const int v = blockIdx.x, n = threadIdx.x;
    float sa = 0.f, sc = 0.f;
    for (int s = 0; s < SLICES_; ++s) {
        sa += wsA[(s * V_ + v) * N_ + n];
        sc += wsC[(s * V_ + v) * N_ + n];
    }
    out[v * N_ + n] = sa + ba[0];
    red[n] = sc + bc[0];
    __syncthreads();
    for (int st = 64; st > 0; st >>= 1) {
        if (n < st) red[n] += red[n + st];
        __syncthreads();
    }
    if (n == 0) out[V_ * N_ + v] = red[0] * (1.f / 128.f);
}

// ---------------------------------------------------------------------------
extern "C" void kernel_launch(void* const* d_in, const int* in_sizes, int n_in,
                              void* d_out, int out_size, void* d_ws, size_t ws_size,
                              hipStream_t stream)
{
    const float* x  = (const float*)d_in[0];
    const float* W1 = (const float*)d_in[1];
    const float* b1 = (const float*)d_in[2];
    const float* W2 = (const float*)d_in[3];
    const float* b2 = (const float*)d_in[4];
    const float* Wa = (const float*)d_in[5];
    const float* ba = (const float*)d_in[6];
    const float* Wc = (const float*)d_in[7];
    const float* bc = (const float*)d_in[8];

    float* wsA = (float*)d_ws;                    // [SLICES][V][N]
    float* wsC = wsA + SLICES_ * V_ * N_;         // [SLICES][V][N]
    float* out = (float*)d_out;                   // actor [V,N] ++ critic [V]

    dim3 grid1(N_ / 16, V_, SLICES_);             // 8 x 8 x 5 = 320 workgroups
    blockmlp_l1<<<grid1, 256, 0, stream>>>(x, W1, b1, W2, b2, Wa, Wc, wsA, wsC);
    blockmlp_head<<<V_, 128, 0, stream>>>(wsA, wsC, ba, bc, out);
}